// SemanticTripletPotential_81509889343868
// MI455X (gfx1250) — compile-verified
//
#include <hip/hip_runtime.h>
#include <hip/hip_bf16.h>
#include <stdint.h>

// Hyperbolic triplet energy + gradient for MI455X (gfx1250, wave32).
//
// Memory-bound gather/scatter: x (102.4 MB) is L2-resident (192 MB L2).
// Per edge we need 8 rows of 1KB ({u, v, 6 candidates}) -> exactly one TDM
// gather-mode descriptor (32-bit index mode = up to 8 row indices, each row
// height=1, width=tile_dim0). TDM loads are double-buffered into LDS per
// wave and tracked with TENSORcnt. WMMA is inapplicable: arithmetic
// intensity is ~0.5 FLOP/byte and no operand is shared across 16 rows/cols,
// so matrix units cannot be fed without >94% wasted MACs; the VALU dot
// products are ~3 orders of magnitude below the compute roofline anyway.

#if __has_include(<hip/amd_detail/amd_gfx1250_TDM.h>)
#define TDM_ARITY6 1   // therock/clang-23 toolchain: 6-arg builtin
#else
#define TDM_ARITY6 0   // ROCm 7.2 / clang-22: 5-arg builtin
#endif

namespace {

typedef unsigned int u32x4 __attribute__((ext_vector_type(4)));
typedef int          i32x4 __attribute__((ext_vector_type(4)));
typedef int          i32x8 __attribute__((ext_vector_type(8)));

constexpr int   kDim           = 256;
constexpr int   kNCand         = 6;
constexpr int   kRows          = 8;          // u, v, w0..w5
constexpr float kMargin        = 0.1f;
constexpr float kEps           = 1e-7f;
constexpr int   kWavesPerBlock = 4;
constexpr int   kBlockSize     = 32 * kWavesPerBlock;
constexpr int   kBufFloats     = kRows * kDim;    // 2048 floats = 8KB / buffer
constexpr int   kWaveFloats    = 2 * kBufFloats;  // double buffered = 16KB/wave

// ---- TDM gather descriptor (loop-invariant part precomputed once) ---------
struct TdmDesc {
  u32x4 g0;   // control | lds_addr | global_addr | type
  i32x8 g1;   // data_size, tensor dims/strides, tile dims
};

__device__ __forceinline__ TdmDesc tdm_make_desc(const float* x, int n) {
  const uint64_t ga = (uint64_t)(uintptr_t)x;
  TdmDesc d;
  d.g0[0] = 0xC0000001u;                  // gather_mode=1, 32-bit idx, count=1
  d.g0[1] = 0u;                           // lds_addr patched per issue
  d.g0[2] = (uint32_t)ga;                 // global_addr[31:0]
  d.g0[3] = (uint32_t)((ga >> 32) & 0x01FFFFFFu) | (2u << 30);  // [56:32]|type=2

  d.g1[0] = (int)(2u << 16);                                    // data_size=4B
  d.g1[1] = (int)(((uint32_t)kDim & 0xFFFFu) << 16);            // tensor_dim0 lo16
  d.g1[2] = (int)(((uint32_t)kDim >> 16) |
                  (((uint32_t)n & 0xFFFFu) << 16));             // dim0 hi | dim1 lo
  d.g1[3] = (int)(((uint32_t)n >> 16) | ((uint32_t)kDim << 16)); // dim1 hi | tile_dim0
  d.g1[4] = kRows;                        // tile_dim1 = #valid gather indices
  d.g1[5] = kDim;                         // tensor_dim0_stride (elements)
  d.g1[6] = 0;
  d.g1[7] = 0;
  return d;
}

// Issue one TDM gather: 8 rows of kDim f32 -> LDS (row r at ldsAddr+r*1KB).
__device__ __forceinline__ void tdm_gather_rows(
    TdmDesc d, uint32_t ldsAddr,
    int r0, int r1, int r2, int r3, int r4, int r5, int r6, int r7) {
  d.g0[1] = ldsAddr;
  i32x4 g2 = { r0, r1, r2, r3 };          // row_index_0..3 (32-bit mode)
  i32x4 g3 = { r4, r5, r6, r7 };          // row_index_4..7
#if TDM_ARITY6
  i32x8 g4 = { 0, 0, 0, 0, 0, 0, 0, 0 };
  __builtin_amdgcn_tensor_load_to_lds(d.g0, d.g1, g2, g3, g4, 0);
#else
  __builtin_amdgcn_tensor_load_to_lds(d.g0, d.g1, g2, g3, 0);
#endif
}

__device__ __forceinline__ float dot8(const float4& a0, const float4& a1,
                                      const float4& b0, const float4& b1) {
  float s = a0.x * b0.x;
  s = fmaf(a0.y, b0.y, s);
  s = fmaf(a0.z, b0.z, s);
  s = fmaf(a0.w, b0.w, s);
  s = fmaf(a1.x, b1.x, s);
  s = fmaf(a1.y, b1.y, s);
  s = fmaf(a1.z, b1.z, s);
  s = fmaf(a1.w, b1.w, s);
  return s;
}

__device__ __forceinline__ void atomic_add4(float* __restrict__ p, float4 v) {
  atomicAdd(p + 0, v.x);
  atomicAdd(p + 1, v.y);
  atomicAdd(p + 2, v.z);
  atomicAdd(p + 3, v.w);
}

// ---- main edge kernel (emitted first so the disasm snippet shows the TDM
// prologue: descriptor build -> tensor_load_to_lds -> s_wait_tensorcnt) -----
__launch_bounds__(kBlockSize)
__global__ void triplet_energy_grad_kernel(
    const float* __restrict__ x,
    const int*   __restrict__ edges,
    const int*   __restrict__ wcand,
    float*       __restrict__ out,   // out[0]=energy, out[1..]=grad(N,256)
    int E, int N) {
  const int lane  = threadIdx.x & 31;
  const int wave  = threadIdx.x >> 5;
  const int wgid  = blockIdx.x * kWavesPerBlock + wave;
  const int eStep = gridDim.x * kWavesPerBlock;
  float* __restrict__ grad = out + 1;

  __shared__ __align__(16) float lbuf[kWavesPerBlock * kWaveFloats];  // 64KB
  float* __restrict__ wbuf = &lbuf[wave * kWaveFloats];
  const uint32_t ldsBase =
      (uint32_t)__builtin_amdgcn_readfirstlane((int)(uintptr_t)wbuf);

  const TdmDesc desc = tdm_make_desc(x, N);

  float acc = 0.0f;   // replicated across lanes of the wave

  // ---- prologue: indices for first edge + launch its TDM gather ----------
  int e = wgid;
  int ui = 0, vi = 0, wi[kNCand] = {};
  if (e < E) {
    ui = __builtin_amdgcn_readfirstlane(edges[2 * e + 0]);
    vi = __builtin_amdgcn_readfirstlane(edges[2 * e + 1]);
#pragma unroll
    for (int c = 0; c < kNCand; ++c)
      wi[c] = __builtin_amdgcn_readfirstlane(wcand[e * kNCand + c]);
    tdm_gather_rows(desc, ldsBase, ui, vi, wi[0], wi[1], wi[2], wi[3], wi[4], wi[5]);
  }

  int buf = 0;
  for (; e < E; e += eStep) {
    // ---- issue next edge's TDM gather into the other buffer --------------
    const int e2 = e + eStep;
    int uiN = 0, viN = 0, wiN[kNCand] = {};
    if (e2 < E) {
      uiN = __builtin_amdgcn_readfirstlane(edges[2 * e2 + 0]);
      viN = __builtin_amdgcn_readfirstlane(edges[2 * e2 + 1]);
#pragma unroll
      for (int c = 0; c < kNCand; ++c)
        wiN[c] = __builtin_amdgcn_readfirstlane(wcand[e2 * kNCand + c]);
      tdm_gather_rows(desc, ldsBase + (uint32_t)((buf ^ 1) * kBufFloats * 4),
                      uiN, viN, wiN[0], wiN[1], wiN[2], wiN[3], wiN[4], wiN[5]);
      __builtin_amdgcn_s_wait_tensorcnt(1);  // oldest (current buf) complete
    } else {
      __builtin_amdgcn_s_wait_tensorcnt(0);
    }

    // prefetch index data 2 iterations ahead (gfx1250 global_prefetch_b8)
    const int e3 = e + 2 * eStep;
    if (e3 < E && lane == 0) {
      __builtin_prefetch(&edges[2 * e3], 0, 2);
      __builtin_prefetch(&wcand[e3 * kNCand], 0, 2);
    }

    // ---- compute from LDS: lane owns elems [4L,4L+4) and [128+4L,...) ----
    const float* rows = wbuf + buf * kBufFloats;
    const float4* __restrict__ u4 = reinterpret_cast<const float4*>(rows);
    const float4* __restrict__ v4 = reinterpret_cast<const float4*>(rows + kDim);
    const float4 ua = u4[lane], ub = u4[lane + 32];
    const float4 va = v4[lane], vb = v4[lane + 32];

    // Minkowski J = diag(-1,1,...,1): global element 0 = lane 0, ua.x
    float4 jua = ua;
    if (lane == 0) jua.x = -jua.x;

    float r[1 + kNCand];
    r[0] = dot8(jua, ub, va, vb);
#pragma unroll
    for (int c = 0; c < kNCand; ++c) {
      const float4* __restrict__ w4 =
          reinterpret_cast<const float4*>(rows + (2 + c) * kDim);
      const float4 wa = w4[lane], wb = w4[lane + 32];
      r[1 + c] = dot8(jua, ub, wa, wb);
    }

    // butterfly reduce across the wave32 -> all lanes hold full sums
#pragma unroll
    for (int m = 16; m >= 1; m >>= 1) {
#pragma unroll
      for (int k = 0; k < 1 + kNCand; ++k)
        r[k] += __shfl_xor(r[k], m, 32);
    }

    const float clampV = -1.0f - kEps;
    const float ipPos  = fminf(r[0], clampV);
    const float dPos   = acoshf(-ipPos);

    int   cBest   = 0;
    float rawBest = r[1];
    float dNeg    = acoshf(-fminf(r[1], clampV));
#pragma unroll
    for (int c = 1; c < kNCand; ++c) {
      const float dc = acoshf(-fminf(r[1 + c], clampV));
      if (dc < dNeg) { dNeg = dc; cBest = c; rawBest = r[1 + c]; }
    }

    const float loss = dPos - dNeg + kMargin;
    if (loss > 0.0f) {
      acc += loss;  // K_SCALE = 1

      // d acosh(-ip)/d ip = -1/sqrt(ip^2-1); clamp gates the gradient
      const float aPos = (r[0] < clampV)
          ? 1.0f / sqrtf(fmaf(ipPos, ipPos, -1.0f)) : 0.0f;
      const float aNeg = (rawBest < clampV)
          ? 1.0f / sqrtf(fmaf(rawBest, rawBest, -1.0f)) : 0.0f;

      // reload argmin candidate row from LDS (cBest is wave-uniform)
      const float4* __restrict__ ws4 =
          reinterpret_cast<const float4*>(rows + (2 + cBest) * kDim);
      const float4 wsa = ws4[lane], wsb = ws4[lane + 32];
      int wSel = wi[0];
#pragma unroll
      for (int c = 1; c < kNCand; ++c)
        if (cBest == c) wSel = wi[c];

      float* __restrict__ gu = grad + (size_t)ui   * kDim;
      float* __restrict__ gv = grad + (size_t)vi   * kDim;
      float* __restrict__ gw = grad + (size_t)wSel * kDim;
      const int ja = 4 * lane;
      const int jb = 128 + 4 * lane;

      if (aPos != 0.0f || aNeg != 0.0f) {
        // grad_u = J o (-aPos * v + aNeg * w*)
        float4 ta, tb;
        ta.x = fmaf(aNeg, wsa.x, -aPos * va.x);
        ta.y = fmaf(aNeg, wsa.y, -aPos * va.y);
        ta.z = fmaf(aNeg, wsa.z, -aPos * va.z);
        ta.w = fmaf(aNeg, wsa.w, -aPos * va.w);
        tb.x = fmaf(aNeg, wsb.x, -aPos * vb.x);
        tb.y = fmaf(aNeg, wsb.y, -aPos * vb.y);
        tb.z = fmaf(aNeg, wsb.z, -aPos * vb.z);
        tb.w = fmaf(aNeg, wsb.w, -aPos * vb.w);
        if (lane == 0) ta.x = -ta.x;  // J on element 0
        atomic_add4(gu + ja, ta);
        atomic_add4(gu + jb, tb);
      }
      if (aPos != 0.0f) {
        // grad_v = -aPos * (J o u)
        float4 sa, sb;
        sa.x = -aPos * ua.x; sa.y = -aPos * ua.y;
        sa.z = -aPos * ua.z; sa.w = -aPos * ua.w;
        sb.x = -aPos * ub.x; sb.y = -aPos * ub.y;
        sb.z = -aPos * ub.z; sb.w = -aPos * ub.w;
        if (lane == 0) sa.x = -sa.x;
        atomic_add4(gv + ja, sa);
        atomic_add4(gv + jb, sb);
      }
      if (aNeg != 0.0f) {
        // grad_w* = aNeg * (J o u)
        float4 qa, qb;
        qa.x = aNeg * ua.x; qa.y = aNeg * ua.y;
        qa.z = aNeg * ua.z; qa.w = aNeg * ua.w;
        qb.x = aNeg * ub.x; qb.y = aNeg * ub.y;
        qb.z = aNeg * ub.z; qb.w = aNeg * ub.w;
        if (lane == 0) qa.x = -qa.x;
        atomic_add4(gw + ja, qa);
        atomic_add4(gw + jb, qb);
      }
    }

    // rotate pipeline state
    ui = uiN; vi = viN;
#pragma unroll
    for (int c = 0; c < kNCand; ++c) wi[c] = wiN[c];
    buf ^= 1;
  }

  // ---- block-level energy reduction: 1 atomic per block (reuse lbuf) -----
  __syncthreads();
  if (lane == 0) lbuf[wave * kWaveFloats] = acc;
  __syncthreads();
  if (threadIdx.x == 0) {
    float s = 0.0f;
#pragma unroll
    for (int i = 0; i < kWavesPerBlock; ++i) s += lbuf[i * kWaveFloats];
    if (s != 0.0f) atomicAdd(out, s);
  }
}

__global__ void zero_out_kernel(float* __restrict__ out, int n) {
  const int tid    = blockIdx.x * blockDim.x + threadIdx.x;
  const int stride = gridDim.x * blockDim.x;
  const int n4     = n >> 2;
  float4* __restrict__ out4 = reinterpret_cast<float4*>(out);
  for (int j = tid; j < n4; j += stride)
    out4[j] = make_float4(0.f, 0.f, 0.f, 0.f);
  for (int j = (n4 << 2) + tid; j < n; j += stride)
    out[j] = 0.0f;
}

}  // namespace

extern "C" void kernel_launch(void* const* d_in, const int* in_sizes, int n_in,
                              void* d_out, int out_size, void* d_ws, size_t ws_size,
                              hipStream_t stream) {
  (void)n_in; (void)d_ws; (void)ws_size;

  const float* x     = (const float*)d_in[0];   // (N, 256) f32
  const int*   edges = (const int*)d_in[1];     // (E, 2)   i32
  const int*   wcand = (const int*)d_in[2];     // (E, 6)   i32
  float*       out   = (float*)d_out;           // [energy, grad(N,256)]

  const int N = in_sizes[0] / kDim;
  const int E = in_sizes[1] / 2;

  // 1) zero energy + grad accumulators (harness does not re-poison between
  //    replays; we accumulate with atomics, so re-zero every call)
  zero_out_kernel<<<2048, 256, 0, stream>>>(out, out_size);

  // 2) edge kernel: one wave32 per edge slot, TDM double-buffered gather
  const int blocksNeeded = (E + kWavesPerBlock - 1) / kWavesPerBlock;
  const int blocks = blocksNeeded < 6144 ? blocksNeeded : 6144;
  triplet_energy_grad_kernel<<<blocks, kBlockSize, 0, stream>>>(
      x, edges, wcand, out, E, N);
}